// GAT_2534030704732
// MI455X (gfx1250) — compile-verified
//
#include <hip/hip_runtime.h>

// ---------------------------------------------------------------------------
// 4-layer GAT for MI455X (gfx1250, wave32).
// GEMMs use V_WMMA_F32_16X16X4_F32 (f32 fidelity; GEMMs are not the bottleneck).
// Edge phase: wave-per-edge coalesced gather + native f32 atomics (L2-resident
// since h/out are 25.6MB each vs 192MB L2).
// ---------------------------------------------------------------------------

typedef __attribute__((ext_vector_type(2))) float v2f;
typedef __attribute__((ext_vector_type(8))) float v8f;

#define HO 64
#define NEG_SLOPE 0.2f

// ---- order-preserving float<->int encoding for atomic segment-max ----------
__device__ __forceinline__ int enc_ord(float f) {
  int i = __float_as_int(f);
  return i >= 0 ? i : (i ^ 0x7FFFFFFF);
}
__device__ __forceinline__ float dec_ord(int k) {
  return __int_as_float(k >= 0 ? k : (k ^ 0x7FFFFFFF));
}

// ---- fills -----------------------------------------------------------------
__global__ void fill_f32(float* __restrict__ p, float v, long long n) {
  long long i = (long long)blockIdx.x * blockDim.x + threadIdx.x;
  if (i < n) p[i] = v;
}
__global__ void fill_i32(int* __restrict__ p, int v, long long n) {
  long long i = (long long)blockIdx.x * blockDim.x + threadIdx.x;
  if (i < n) p[i] = v;
}

// ---- h = x @ W  via WMMA f32 16x16x4 --------------------------------------
// block = 256 threads = 8 waves; each wave owns a 16x16 tile of h.
// Block covers 32 rows x 64 cols. W staged in LDS.
__global__ __launch_bounds__(256) void gat_gemm_wmma(const float* __restrict__ x,
                                                     const float* __restrict__ W,
                                                     float* __restrict__ h,
                                                     int N, int Fin) {
  __shared__ float Wlds[128 * HO];  // up to 32KB (Fin<=128)
  const int tid = threadIdx.x;
  for (int i = tid; i < Fin * HO; i += 256) Wlds[i] = W[i];
  __syncthreads();

  const int wave = tid >> 5;
  const int lane = tid & 31;
  const int hi   = lane >> 4;   // 0: K+0/K+1  1: K+2/K+3
  const int mr   = lane & 15;
  const int rowbase = blockIdx.x * 32 + (wave >> 2) * 16;
  const int colbase = (wave & 3) * 16;

  int row = rowbase + mr;
  if (row >= N) row = N - 1;                 // keep EXEC all-ones for WMMA
  const float* xr = x + (size_t)row * Fin;

  v8f acc = {};
  for (int k0 = 0; k0 < Fin; k0 += 4) {
    const int ka = k0 + 2 * hi;
    v2f a, b;
    a.x = xr[ka];
    a.y = xr[ka + 1];
    b.x = Wlds[ka * HO + colbase + mr];
    b.y = Wlds[(ka + 1) * HO + colbase + mr];
    acc = __builtin_amdgcn_wmma_f32_16x16x4_f32(false, a, false, b,
                                                (short)0, acc, false, false);
  }

#pragma unroll
  for (int v = 0; v < 8; ++v) {
    const int r = rowbase + v + 8 * hi;      // C/D layout: M = v + 8*hi
    if (r < N) h[(size_t)r * HO + colbase + mr] = acc[v];
  }
}

// ---- per-node attention dots: es = h.a_s, ed = h.a_d (wave per node) ------
__global__ __launch_bounds__(256) void gat_node_dots(const float* __restrict__ h,
                                                     const float* __restrict__ a_s,
                                                     const float* __restrict__ a_d,
                                                     float* __restrict__ es,
                                                     float* __restrict__ ed, int N) {
  const int node = (blockIdx.x * blockDim.x + threadIdx.x) >> 5;
  const int lane = threadIdx.x & 31;
  if (node >= N) return;
  const float* hr = h + (size_t)node * HO;
  float h0 = hr[lane], h1 = hr[lane + 32];
  float s0 = h0 * a_s[lane] + h1 * a_s[lane + 32];
  float s1 = h0 * a_d[lane] + h1 * a_d[lane + 32];
#pragma unroll
  for (int off = 16; off > 0; off >>= 1) {
    s0 += __shfl_xor(s0, off, 32);
    s1 += __shfl_xor(s1, off, 32);
  }
  if (lane == 0) { es[node] = s0; ed[node] = s1; }
}

// ---- edge src/dst (self-loops appended implicitly) -------------------------
__device__ __forceinline__ void edge_sd(const int* __restrict__ ei, int E, int i,
                                        int& s, int& d) {
  if (i < E) { s = ei[i]; d = ei[E + i]; }
  else       { s = d = i - E; }
}

// ---- edge logits: e = leaky_relu(es[s]+ed[d]); segment max via atomicMax ---
__global__ __launch_bounds__(256) void gat_edge_logits(const int* __restrict__ ei, int E, int ET,
                                                       const float* __restrict__ es,
                                                       const float* __restrict__ ed,
                                                       float* __restrict__ e,
                                                       int* __restrict__ mEnc) {
  const int i = blockIdx.x * blockDim.x + threadIdx.x;
  if (i >= ET) return;
  int s, d; edge_sd(ei, E, i, s, d);
  float v = es[s] + ed[d];
  v = v > 0.f ? v : NEG_SLOPE * v;
  e[i] = v;
  atomicMax(&mEnc[d], enc_ord(v));
}

// ---- edge exp: e <- exp(e - m[d]); denom[d] += e ---------------------------
__global__ __launch_bounds__(256) void gat_edge_exp(const int* __restrict__ ei, int E, int ET,
                                                    float* __restrict__ e,
                                                    const int* __restrict__ mEnc,
                                                    float* __restrict__ denom) {
  const int i = blockIdx.x * blockDim.x + threadIdx.x;
  if (i >= ET) return;
  int s, d; edge_sd(ei, E, i, s, d);
  const float ex = expf(e[i] - dec_ord(mEnc[d]));
  e[i] = ex;
  atomicAdd(&denom[d], ex);
}

// ---- aggregation (64 feats): wave per edge, 2 feats/lane -------------------
__global__ __launch_bounds__(256) void gat_aggregate64(const int* __restrict__ ei, int E, int ET,
                                                       const float* __restrict__ h,
                                                       const float* __restrict__ e,
                                                       const float* __restrict__ denom,
                                                       float* __restrict__ out) {
  const long long gid = (long long)blockIdx.x * blockDim.x + threadIdx.x;
  const long long w = gid >> 5;
  const int lane = threadIdx.x & 31;
  if (w >= ET) return;
  const int i = (int)w;
  int s, d; edge_sd(ei, E, i, s, d);
  const float alpha = e[i] / (denom[d] + 1e-16f);
  const float* hr = h + (size_t)s * HO;
  float* o = out + (size_t)d * HO;
  atomicAdd(&o[lane],      hr[lane]      * alpha);
  atomicAdd(&o[lane + 32], hr[lane + 32] * alpha);
}

// ---- out = relu(out + b) ---------------------------------------------------
__global__ void gat_bias_relu(float* __restrict__ out, const float* __restrict__ b,
                              long long n) {
  const long long i = (long long)blockIdx.x * blockDim.x + threadIdx.x;
  if (i >= n) return;
  const float v = out[i] + b[i & 63];
  out[i] = v > 0.f ? v : 0.f;
}

// ---- final layer: hl = x @ Wl (F_out=1); es/ed scalars (wave per node) -----
__global__ __launch_bounds__(256) void gat_final_node(const float* __restrict__ x,
                                                      const float* __restrict__ Wl,
                                                      const float* __restrict__ asl,
                                                      const float* __restrict__ adl,
                                                      float* __restrict__ hl,
                                                      float* __restrict__ es,
                                                      float* __restrict__ ed, int N) {
  const int node = (blockIdx.x * blockDim.x + threadIdx.x) >> 5;
  const int lane = threadIdx.x & 31;
  if (node >= N) return;
  const float* xr = x + (size_t)node * HO;
  float s = xr[lane] * Wl[lane] + xr[lane + 32] * Wl[lane + 32];
#pragma unroll
  for (int off = 16; off > 0; off >>= 1) s += __shfl_xor(s, off, 32);
  if (lane == 0) { hl[node] = s; es[node] = s * asl[0]; ed[node] = s * adl[0]; }
}

// ---- final scalar aggregation ----------------------------------------------
__global__ __launch_bounds__(256) void gat_aggregate1(const int* __restrict__ ei, int E, int ET,
                                                      const float* __restrict__ hl,
                                                      const float* __restrict__ e,
                                                      const float* __restrict__ denom,
                                                      float* __restrict__ out) {
  const int i = blockIdx.x * blockDim.x + threadIdx.x;
  if (i >= ET) return;
  int s, d; edge_sd(ei, E, i, s, d);
  const float alpha = e[i] / (denom[d] + 1e-16f);
  atomicAdd(&out[d], hl[s] * alpha);
}

__global__ void gat_add_bias1(float* __restrict__ out, const float* __restrict__ bl, int n) {
  const int i = blockIdx.x * blockDim.x + threadIdx.x;
  if (i < n) out[i] += bl[0];
}

// ---------------------------------------------------------------------------
static inline int cdiv(long long a, long long b) { return (int)((a + b - 1) / b); }

extern "C" void kernel_launch(void* const* d_in, const int* in_sizes, int n_in,
                              void* d_out, int out_size, void* d_ws, size_t ws_size,
                              hipStream_t stream) {
  const float* x0 = (const float*)d_in[0];
  const int*   ei = (const int*)d_in[1];     // [2,E] flat int32, values < N
  const int N  = in_sizes[0] / 128;
  const int E  = in_sizes[1] / 2;
  const int ET = E + N;                      // edges + self-loops

  // workspace carve (256B aligned)
  char* p = (char*)d_ws;
  auto carve = [&](size_t bytes) {
    void* r = (void*)p;
    p += (bytes + 255) & ~(size_t)255;
    return r;
  };
  float* h     = (float*)carve((size_t)N * HO * 4);
  float* xA    = (float*)carve((size_t)N * HO * 4);
  float* xB    = (float*)carve((size_t)N * HO * 4);
  float* es    = (float*)carve((size_t)N * 4);
  float* ed    = (float*)carve((size_t)N * 4);
  int*   mEnc  = (int*)  carve((size_t)N * 4);
  float* denom = (float*)carve((size_t)N * 4);
  float* e     = (float*)carve((size_t)ET * 4);
  float* hl    = (float*)carve((size_t)N * 4);

  const int NEG_INF_ENC = (int)(0xFF800000u ^ 0x7FFFFFFFu);  // enc(-inf)

  const float* xin = x0;
  int Fin = 128;
  float* outs[3] = { xA, xB, xA };

  for (int l = 0; l < 3; ++l) {
    const float* W  = (const float*)d_in[3 + l * 4];
    const float* as = (const float*)d_in[4 + l * 4];
    const float* ad = (const float*)d_in[5 + l * 4];
    const float* b  = (const float*)d_in[6 + l * 4];
    float* out = outs[l];

    gat_gemm_wmma<<<cdiv(N, 32), 256, 0, stream>>>(xin, W, h, N, Fin);
    gat_node_dots<<<cdiv((long long)N * 32, 256), 256, 0, stream>>>(h, as, ad, es, ed, N);

    fill_i32<<<cdiv(N, 256), 256, 0, stream>>>(mEnc, NEG_INF_ENC, N);
    fill_f32<<<cdiv(N, 256), 256, 0, stream>>>(denom, 0.f, N);
    fill_f32<<<cdiv((long long)N * HO, 256), 256, 0, stream>>>(out, 0.f, (long long)N * HO);

    gat_edge_logits<<<cdiv(ET, 256), 256, 0, stream>>>(ei, E, ET, es, ed, e, mEnc);
    gat_edge_exp<<<cdiv(ET, 256), 256, 0, stream>>>(ei, E, ET, e, mEnc, denom);
    gat_aggregate64<<<cdiv((long long)ET * 32, 256), 256, 0, stream>>>(ei, E, ET, h, e, denom, out);
    gat_bias_relu<<<cdiv((long long)N * HO, 256), 256, 0, stream>>>(out, b, (long long)N * HO);

    xin = out;
    Fin = HO;
  }

  // final layer (F_out = 1)
  const float* Wl  = (const float*)d_in[15];
  const float* asl = (const float*)d_in[16];
  const float* adl = (const float*)d_in[17];
  const float* bl  = (const float*)d_in[18];
  float* outF = (float*)d_out;

  gat_final_node<<<cdiv((long long)N * 32, 256), 256, 0, stream>>>(xin, Wl, asl, adl, hl, es, ed, N);

  fill_i32<<<cdiv(N, 256), 256, 0, stream>>>(mEnc, NEG_INF_ENC, N);
  fill_f32<<<cdiv(N, 256), 256, 0, stream>>>(denom, 0.f, N);
  fill_f32<<<cdiv(N, 256), 256, 0, stream>>>(outF, 0.f, N);

  gat_edge_logits<<<cdiv(ET, 256), 256, 0, stream>>>(ei, E, ET, es, ed, e, mEnc);
  gat_edge_exp<<<cdiv(ET, 256), 256, 0, stream>>>(ei, E, ET, e, mEnc, denom);
  gat_aggregate1<<<cdiv(ET, 256), 256, 0, stream>>>(ei, E, ET, hl, e, denom, outF);
  gat_add_bias1<<<cdiv(N, 256), 256, 0, stream>>>(outF, bl, N);
}